// HashEmbeddingBag_66331474919971
// MI455X (gfx1250) — compile-verified
//
#include <hip/hip_runtime.h>

// HashEmbeddingBag mean on MI455X (gfx1250, wave32).
// One wave per bag. Masked sum via V_WMMA_F32_16X16X4_F32:
//   D(16x16) = A(16x4 mask) x B(4x16 gathered dims) + C
// Mask rows of A are M-invariant, so every D row holds the tile's partial sum.
// Memory-bound: ~105 MB of random 256B row gathers vs 23.3 TB/s HBM; WMMA cost
// is negligible, it just fuses mask+accumulate branch-free. Prefetch index is
// software-pipelined (and pinned so the compiler can't sink the load into the
// prefetch block), so the prefetch path carries no dependent s_wait_loadcnt.

typedef __attribute__((ext_vector_type(2))) float v2f;
typedef __attribute__((ext_vector_type(8))) float v8f;

#define EMB_L 200
#define EMB_D 64

__device__ __forceinline__ int clampL(int l) { return l < EMB_L ? l : (EMB_L - 1); }

__global__ __launch_bounds__(256)
void hash_embedding_bag_wmma(const float* __restrict__ weight,
                             const long long* __restrict__ idx,
                             const long long* __restrict__ lengths,
                             float* __restrict__ out,
                             int B) {
  const int lane = (int)threadIdx.x & 31;
  const int bag  = (int)blockIdx.x * 8 + ((int)threadIdx.x >> 5);
  if (bag >= B) return;  // wave-uniform

  // View of the low dwords of the int64 indices (values < 2^20, little-endian).
  const int* __restrict__ bidx = (const int*)(idx + (long long)bag * EMB_L);
  // Force bag length into an SGPR: scalar loop control, EXEC stays all-1s.
  const int n = __builtin_amdgcn_readfirstlane((int)lengths[bag]);

  // Lane roles per ISA f32 16x16x4 operand layouts (wave32):
  //  A: lanes 0-15 -> K={0,1}, lanes 16-31 -> K={2,3}
  //  B: reg0 = row k0 (N = lane&15), reg1 = row k0+1
  const int half = lane >> 4;
  const int col  = lane & 15;
  const int k0   = half * 2;
  const int k1   = k0 + 1;

  // Prefetch pattern: 32 lanes cover 8 rows (4 lanes/row, 64B apart).
  const int pfRow = lane >> 2;        // 0..7
  const int pfOff = (lane & 3) * 16;  // in floats (64B)

  v8f c0 = {}; v8f c1 = {}; v8f c2 = {}; v8f c3 = {};

  // Software-pipelined prefetch index: value consumed at iteration l covers
  // rows l+8..l+15 and was loaded during iteration l-8.
  int pfi = bidx[2 * clampL(8 + pfRow)];

  for (int l = 0; l < n; l += 8) {
    // Prefetch rows l+8..l+15 (index already resident -> no dependent wait).
    if (l + 8 < n)  // scalar guard
      __builtin_prefetch(weight + pfi * EMB_D + pfOff, 0, 3);
    // Issue next prefetch-index load; completion is covered by the
    // s_wait_loadcnt 0 the final WMMA of this iteration already requires.
    pfi = bidx[2 * clampL(l + 16 + pfRow)];

    // ---- rows l .. l+3 ----
    const int la0 = l + k0, la1 = l + k1;
    const int ia0 = bidx[2 * clampL(la0)];
    const int ia1 = bidx[2 * clampL(la1)];
    v2f aA;
    aA[0] = (la0 < n) ? 1.0f : 0.0f;
    aA[1] = (la1 < n) ? 1.0f : 0.0f;
    const float* pa0 = weight + ia0 * EMB_D + col;
    const float* pa1 = weight + ia1 * EMB_D + col;
    v2f bA0; bA0[0] = pa0[0];  bA0[1] = pa1[0];
    v2f bA1; bA1[0] = pa0[16]; bA1[1] = pa1[16];
    v2f bA2; bA2[0] = pa0[32]; bA2[1] = pa1[32];
    v2f bA3; bA3[0] = pa0[48]; bA3[1] = pa1[48];

    // ---- rows l+4 .. l+7 ----
    const int lb0 = l + 4 + k0, lb1 = l + 4 + k1;
    const int ib0 = bidx[2 * clampL(lb0)];
    const int ib1 = bidx[2 * clampL(lb1)];
    v2f aB;
    aB[0] = (lb0 < n) ? 1.0f : 0.0f;
    aB[1] = (lb1 < n) ? 1.0f : 0.0f;
    const float* pb0 = weight + ib0 * EMB_D + col;
    const float* pb1 = weight + ib1 * EMB_D + col;
    v2f bB0; bB0[0] = pb0[0];  bB0[1] = pb1[0];
    v2f bB1; bB1[0] = pb0[16]; bB1[1] = pb1[16];
    v2f bB2; bB2[0] = pb0[32]; bB2[1] = pb1[32];
    v2f bB3; bB3[0] = pb0[48]; bB3[1] = pb1[48];

    // 8 args: (neg_a, A, neg_b, B, c_mod, C, reuse_a, reuse_b)
    c0 = __builtin_amdgcn_wmma_f32_16x16x4_f32(false, aA, false, bA0, (short)0, c0, false, false);
    c1 = __builtin_amdgcn_wmma_f32_16x16x4_f32(false, aA, false, bA1, (short)0, c1, false, false);
    c2 = __builtin_amdgcn_wmma_f32_16x16x4_f32(false, aA, false, bA2, (short)0, c2, false, false);
    c3 = __builtin_amdgcn_wmma_f32_16x16x4_f32(false, aA, false, bA3, (short)0, c3, false, false);
    c0 = __builtin_amdgcn_wmma_f32_16x16x4_f32(false, aB, false, bB0, (short)0, c0, false, false);
    c1 = __builtin_amdgcn_wmma_f32_16x16x4_f32(false, aB, false, bB1, (short)0, c1, false, false);
    c2 = __builtin_amdgcn_wmma_f32_16x16x4_f32(false, aB, false, bB2, (short)0, c2, false, false);
    c3 = __builtin_amdgcn_wmma_f32_16x16x4_f32(false, aB, false, bB3, (short)0, c3, false, false);

    // Pin pfi here: forces its load to complete in THIS iteration (already
    // covered by the loadcnt-0 wait above), so the compiler cannot sink the
    // load next to the prefetch and re-introduce a dependent stall.
    asm volatile("" : "+v"(pfi));
  }

  // C/D layout: VGPR0 lanes0-15 = (M=0,N=col); lanes16-31 = (M=8,N=col).
  // All M rows identical -> lanes 0-15 emit tiles 0,1; lanes 16-31 tiles 2,3.
  const float den = (float)(n > 0 ? n : 1);
  float* __restrict__ obag = out + (long long)bag * EMB_D;
  if (half == 0) {
    obag[col]      = c0[0] / den;
    obag[16 + col] = c1[0] / den;
  } else {
    obag[32 + col] = c2[0] / den;
    obag[48 + col] = c3[0] / den;
  }
}

extern "C" void kernel_launch(void* const* d_in, const int* in_sizes, int n_in,
                              void* d_out, int out_size, void* d_ws, size_t ws_size,
                              hipStream_t stream) {
  const float*     weight  = (const float*)d_in[0];
  const long long* idx     = (const long long*)d_in[1];
  const long long* lengths = (const long long*)d_in[2];
  float*           out     = (float*)d_out;

  const int B = in_sizes[2];            // 4096 bags
  const int blocks = (B + 7) / 8;       // 8 waves (bags) per 256-thread block
  hash_embedding_bag_wmma<<<blocks, 256, 0, stream>>>(weight, idx, lengths, out, B);
}